// AggregationLayer_2963527434957
// MI455X (gfx1250) — compile-verified
//
#include <hip/hip_runtime.h>

typedef float v2f __attribute__((ext_vector_type(2)));
typedef float v4f __attribute__((ext_vector_type(4)));
typedef float v8f __attribute__((ext_vector_type(8)));

#define DFEAT 64
#define SEGS_PER_WAVE 16

// One wave32 handles 16 consecutive segments (sorted segment ids => contiguous
// edge range, found by scalar binary searches). Edges are staged 32 at a time:
// one coalesced load of segment ids + gather indices per chunk, then run
// detection and index distribution happen on-chip via ds_bpermute (__shfl).
// Segmented sum = WMMA f32 16x16x4 scatter-add: A = 16 feature rows x 4 edges
// of gathered values (documented A layout, tail-gated per K slot); B = 4x16
// segment-indicator, CONSTANT across K (immune to B's K-layout details).
// All loop control is scalarized via readfirstlane so EXEC stays all-1s at
// every WMMA, per the ISA requirement.
__global__ __launch_bounds__(256) void seg_mean_wmma_kernel(
    const float* __restrict__ values,   // [N_SRC, 64] f32
    const int*   __restrict__ gidx,     // [E]
    const int*   __restrict__ segid,    // [E] sorted
    float*       __restrict__ out,      // [S, 64] f32
    int E, int S)
{
    const int lane      = threadIdx.x & 31;
    const int waveInBlk = __builtin_amdgcn_readfirstlane(threadIdx.x >> 5);
    const int wave      = blockIdx.x * (blockDim.x >> 5) + waveInBlk;   // scalar
    const int base      = wave * SEGS_PER_WAVE;                         // scalar
    if (base >= S) return;   // scalar branch, whole wave exits together

    // --- scalar binary searches for this wave's edge range ---
    int eStart, eEnd;
    {
        int lo = 0, hi = E;
        while (lo < hi) {
            int mid = (lo + hi) >> 1;
            if (segid[mid] < base) lo = mid + 1; else hi = mid;
        }
        eStart = lo;
        const int key = base + SEGS_PER_WAVE;
        hi = E;                        // second search starts at eStart
        while (lo < hi) {
            int mid = (lo + hi) >> 1;
            if (segid[mid] < key) lo = mid + 1; else hi = mid;
        }
        eEnd = lo;
    }
    eStart = __builtin_amdgcn_readfirstlane(eStart);
    eEnd   = __builtin_amdgcn_readfirstlane(eEnd);

    const int m     = lane & 15;          // A: feature row within tile
    const int n     = lane & 15;          // B/D: segment column within wave's 16
    const int kBase = (lane >> 4) << 1;   // A layout: lanes 0-15 K{0,1}, 16-31 K{2,3}

    v8f acc0 = {}, acc1 = {}, acc2 = {}, acc3 = {};
    int myCnt = 0;

    for (int eC = eStart; eC < eEnd; eC += 32) {
        const int cl = __builtin_amdgcn_readfirstlane(min(32, eEnd - eC));
        const int eL = min(eC + lane, eEnd - 1);
        const int sC = segid[eL];         // coalesced b32: 32 segment ids
        const int gC = gidx[eL];          // coalesced b32: 32 gather indices

        int r = 0;
        while (r < cl) {                  // scalar loop control
            // maximal same-segment run of <=4 edges, detected on-chip
            int s0 = __shfl(sC, r, 32);
            int s1 = __shfl(sC, (r + 1 < cl) ? r + 1 : r, 32);
            int s2 = __shfl(sC, (r + 2 < cl) ? r + 2 : r, 32);
            int s3 = __shfl(sC, (r + 3 < cl) ? r + 3 : r, 32);
            int ok1 = ((r + 1) < cl) & (s1 == s0);
            int ok2 = ok1 & (((r + 2) < cl) & (s2 == s0));
            int ok3 = ok2 & (((r + 3) < cl) & (s3 == s0));
            int len = __builtin_amdgcn_readfirstlane(1 + ok1 + ok2 + ok3);
            s0      = __builtin_amdgcn_readfirstlane(s0);

            const int segLocal = s0 - base;               // 0..15, scalar
            const float bv = (n == segLocal) ? 1.0f : 0.0f;
            v2f b; b.x = bv; b.y = bv;                    // constant across K
            myCnt += (n == segLocal) ? len : 0;

            // this lane's two K slots (clamped into the run) and their rows
            const int sa = r + ((kBase     < len) ? kBase     : len - 1);
            const int sb = r + ((kBase + 1 < len) ? kBase + 1 : len - 1);
            const bool gx = (kBase     < len);
            const bool gy = (kBase + 1) < len;
            const int ia = __shfl(gC, sa, 32);            // ds_bpermute
            const int ib = __shfl(gC, sb, 32);
            const float* pa = values + (long)ia * DFEAT + m;
            const float* pb = values + (long)ib * DFEAT + m;

            const float vx0 = pa[0],  vy0 = pb[0];
            const float vx1 = pa[16], vy1 = pb[16];
            const float vx2 = pa[32], vy2 = pb[32];
            const float vx3 = pa[48], vy3 = pb[48];

            v2f a0, a1, a2, a3;
            a0.x = gx ? vx0 : 0.0f;  a0.y = gy ? vy0 : 0.0f;
            a1.x = gx ? vx1 : 0.0f;  a1.y = gy ? vy1 : 0.0f;
            a2.x = gx ? vx2 : 0.0f;  a2.y = gy ? vy2 : 0.0f;
            a3.x = gx ? vx3 : 0.0f;  a3.y = gy ? vy3 : 0.0f;

            acc0 = __builtin_amdgcn_wmma_f32_16x16x4_f32(false, a0, false, b, (short)0, acc0, false, false);
            acc1 = __builtin_amdgcn_wmma_f32_16x16x4_f32(false, a1, false, b, (short)0, acc1, false, false);
            acc2 = __builtin_amdgcn_wmma_f32_16x16x4_f32(false, a2, false, b, (short)0, acc2, false, false);
            acc3 = __builtin_amdgcn_wmma_f32_16x16x4_f32(false, a3, false, b, (short)0, acc3, false, false);

            r += len;
        }
    }

    // --- mean + writeback ---
    float cnt = (float)myCnt;
    if (cnt < 1.0f) cnt = 1.0f;           // empty segments -> 0
    const float inv = 1.0f / cnt;

    // D layout: VGPR r -> M=r (lanes 0-15) or M=r+8 (lanes 16-31), N = lane&15
    const int mRow = (lane < 16) ? 0 : 8;
    float* orow = out + (long)(base + n) * DFEAT + mRow;

#define STORE_TILE(ACC, T)                                              \
    do {                                                                \
        v4f w0, w1;                                                     \
        w0.x = ACC[0] * inv; w0.y = ACC[1] * inv;                       \
        w0.z = ACC[2] * inv; w0.w = ACC[3] * inv;                       \
        w1.x = ACC[4] * inv; w1.y = ACC[5] * inv;                       \
        w1.z = ACC[6] * inv; w1.w = ACC[7] * inv;                       \
        *(v4f*)(orow + (T) * 16)     = w0;                              \
        *(v4f*)(orow + (T) * 16 + 4) = w1;                              \
    } while (0)

    STORE_TILE(acc0, 0);
    STORE_TILE(acc1, 1);
    STORE_TILE(acc2, 2);
    STORE_TILE(acc3, 3);
#undef STORE_TILE
}

extern "C" void kernel_launch(void* const* d_in, const int* in_sizes, int n_in,
                              void* d_out, int out_size, void* d_ws, size_t ws_size,
                              hipStream_t stream) {
    const float* values = (const float*)d_in[0];
    const int*   gidx   = (const int*)d_in[1];
    const int*   segid  = (const int*)d_in[2];
    // d_in[3] (num_segments scalar) lives on device; derive S from out_size.
    float* out = (float*)d_out;

    const int E = in_sizes[1];
    const int S = out_size / DFEAT;

    const int waves         = (S + SEGS_PER_WAVE - 1) / SEGS_PER_WAVE;
    const int wavesPerBlock = 8;                       // 256 threads = 8 wave32
    const int blocks        = (waves + wavesPerBlock - 1) / wavesPerBlock;

    seg_mean_wmma_kernel<<<blocks, wavesPerBlock * 32, 0, stream>>>(
        values, gidx, segid, out, E, S);
}